// CPABActivationSame_53197464928906
// MI455X (gfx1250) — compile-verified
//
#include <hip/hip_runtime.h>
#include <hip/hip_bf16.h>
#include <stdint.h>

// CPAB config (matches reference)
#define CPAB_NC   16
#define CPAB_NS1  10
#define CPAB_NS2  5

__device__ __forceinline__ int cpab_cell(float x) {
    float c = floorf(x * 16.0f);
    c = fminf(fmaxf(c, 0.0f), 15.0f);
    return (int)c;
}

// tabA[c] = {m, k}: psi(x) = m*x + k   (m = exp(dT*a_c) or 1; k = (b/a)(eta-1) or dT*b)
// tabB[c] = {p, q}: one Euler substep  xe = p*xe + q  (p = 1 + h2*a_c, q = h2*b_c)
__device__ __forceinline__ float cpab_one(float x0,
                                          const float2* __restrict__ tabA,
                                          const float2* __restrict__ tabB) {
    float xs = (x0 + 4.0f) * 0.125f;                 // rescale to [0,1]
    bool outside = (xs >= 1.0f) || (xs <= 0.0f);
    float xp = xs;
#pragma unroll
    for (int s = 0; s < CPAB_NS1; ++s) {
        int c0 = cpab_cell(xp);
        float2 mk = tabA[c0];                        // ds_load_b64
        float xt = fmaf(mk.x, xp, mk.y);             // closed-form within-cell step
        if (cpab_cell(xt) != c0) {                   // crossed a boundary: Euler refine
            float xe = xp;
#pragma unroll
            for (int j = 0; j < CPAB_NS2; ++j) {
                int c = cpab_cell(xe);
                float2 pq = tabB[c];                 // ds_load_b64
                xe = fmaf(pq.x, xe, pq.y);
            }
            xt = xe;
        }
        xp = xt;
    }
    float r = fmaf(xp, 8.0f, -4.0f);                 // rescale back
    return outside ? x0 : r;                         // identity outside domain
}

__global__ __launch_bounds__(256)
void cpab_activation_kernel(const float* __restrict__ x,
                            const float* __restrict__ timep,
                            const float* __restrict__ theta,
                            float* __restrict__ out,
                            int n, int ntheta) {
    __shared__ float2 tabA[CPAB_NC];
    __shared__ float2 tabB[CPAB_NC];
    __shared__ float4 tile[256];

    const int tid = threadIdx.x;
    const long long gi = (long long)blockIdx.x * 256 + tid;   // float4 index
    const bool full = (gi * 4 + 3) < (long long)n;

    // --- CDNA5 async path: stage this thread's float4 of x into LDS ---------
    if (full) {
        uint32_t ldsaddr = (uint32_t)(uintptr_t)(void*)&tile[tid]; // low 32b = LDS offset
        uint32_t voff    = (uint32_t)(gi * 16);                    // byte offset (GVS mode)
        asm volatile("global_load_async_to_lds_b128 %0, %1, %2"
                     :
                     : "v"(ldsaddr), "v"(voff), "s"(x)
                     : "memory");
    }

    // --- Build per-cell tables while the async copy is in flight ------------
    if (tid < CPAB_NC) {
        float t  = timep[0];
        float dT = t * (1.0f / CPAB_NS1);
        float h2 = dT * (1.0f / CPAB_NS2);
        int   c  = tid;
        // zero-boundary knot velocities: v = [0, theta..., 0]
        float vc   = (c == 0)           ? 0.0f : theta[c - 1];
        float vc1  = (c == CPAB_NC - 1) ? 0.0f : theta[c];
        float a = (vc1 - vc) * 16.0f;            // slope (h = 1/16)
        float b = vc - a * ((float)c * 0.0625f); // intercept (xk = c/16)
        bool  zero   = fabsf(a) < 1e-8f;
        float a_safe = zero ? 1.0f : a;
        float eta    = expf(dT * a);
        float m = zero ? 1.0f : eta;
        float k = zero ? (dT * b) : (b / a_safe) * (eta - 1.0f);
        tabA[c] = make_float2(m, k);
        tabB[c] = make_float2(1.0f + h2 * a, h2 * b);
    }
    __syncthreads();

    // wait for our async global->LDS copy to land
    asm volatile("s_wait_asynccnt 0" ::: "memory");

    if (full) {
        float4 v = tile[tid];                     // ds_load_b128
        float4 r;
        r.x = cpab_one(v.x, tabA, tabB);
        r.y = cpab_one(v.y, tabA, tabB);
        r.z = cpab_one(v.z, tabA, tabB);
        r.w = cpab_one(v.w, tabA, tabB);
        reinterpret_cast<float4*>(out)[gi] = r;   // global_store_b128
    } else {
        long long base = gi * 4;
        for (int j = 0; j < 4; ++j) {
            long long idx = base + j;
            if (idx < (long long)n) out[idx] = cpab_one(x[idx], tabA, tabB);
        }
    }

    // second tuple output: theta passthrough
    if (blockIdx.x == 0 && tid < ntheta)
        out[(long long)n + tid] = theta[tid];
}

extern "C" void kernel_launch(void* const* d_in, const int* in_sizes, int n_in,
                              void* d_out, int out_size, void* d_ws, size_t ws_size,
                              hipStream_t stream) {
    // setup_inputs order: x, edge_index, edge_attr, batch, time, theta
    const float* x     = (const float*)d_in[0];
    const float* timep = (const float*)d_in[4];
    const float* theta = (const float*)d_in[5];
    float* out = (float*)d_out;

    int n      = in_sizes[0];
    int ntheta = (n_in > 5) ? in_sizes[5] : 15;

    int n4   = (n + 3) / 4;
    int grid = (n4 + 255) / 256;
    cpab_activation_kernel<<<grid, 256, 0, stream>>>(x, timep, theta, out, n, ntheta);
}